// PatternAwareNormalization_79680233275554
// MI455X (gfx1250) — compile-verified
//
#include <hip/hip_runtime.h>
#include <math.h>

typedef __attribute__((ext_vector_type(2))) float v2f;
typedef __attribute__((ext_vector_type(8))) float v8f;

#define B_DIM  16
#define T_DIM  4096
#define D_FEAT 1024
#define NPAT   8
#define PLEN   128            /* T/4/NPAT */
#define M_TOT  (NPAT * PLEN)  /* 1024 gathered rows per batch */

// ---------------------------------------------------------------------------
// Full-wave (32-lane) sum through the f32 WMMA unit.
// With B = all-ones, D[m][n] = sum_k A[m][k]; the all-element sum of A is
// independent of A's lane layout, so each lane contributes its partial in
// a.x (a.y = 0).  Per the CDNA5 C/D layout, a lane's 8 accumulator VGPRs at
// fixed n cover M=0..7 (lanes 0-15) or M=8..15 (lanes 16-31); adding the two
// half-sums (lane 0 + lane 16) yields the total over all 32 lanes.
// EXEC must be all 1s: callers invoke this from fully-active blocks.
// ---------------------------------------------------------------------------
__device__ __forceinline__ float wave_sum_wmma(float v) {
  v2f a; a.x = v;    a.y = 0.0f;
  v2f b; b.x = 1.0f; b.y = 1.0f;
  v8f c = {};
  c = __builtin_amdgcn_wmma_f32_16x16x4_f32(false, a, false, b,
                                            (short)0, c, false, false);
  float r = c[0] + c[1] + c[2] + c[3] + c[4] + c[5] + c[6] + c[7];
  return __shfl(r, 0, 32) + __shfl(r, 16, 32);
}

// ---------------------------------------------------------------------------
// Pass 1: per-row sum and sum-of-squares over the D=1024 channels.
// One block of 256 threads per (b,t) row; float4 coalesced 128-bit loads.
// ---------------------------------------------------------------------------
__global__ __launch_bounds__(256) void k_rowstats(const float* __restrict__ x,
                                                  float* __restrict__ rowsum,
                                                  float* __restrict__ rowsumsq) {
  const int row = blockIdx.x;          // b*T + t
  const int tid = threadIdx.x;
  const float4 v = ((const float4*)(x + (size_t)row * D_FEAT))[tid];
  float s = v.x + v.y + v.z + v.w;
  float q = v.x * v.x + v.y * v.y + v.z * v.z + v.w * v.w;

  const float sw = wave_sum_wmma(s);
  const float qw = wave_sum_wmma(q);

  __shared__ float ls[8], lq[8];
  const int lane = tid & 31, wid = tid >> 5;
  if (lane == 0) { ls[wid] = sw; lq[wid] = qw; }
  __syncthreads();
  if (tid == 0) {
    float S = 0.f, Q = 0.f;
#pragma unroll
    for (int i = 0; i < 8; ++i) { S += ls[i]; Q += lq[i]; }
    rowsum[row]   = S;
    rowsumsq[row] = Q;
  }
}

// Monotone float -> uint mapping for ordered comparison.
__device__ __forceinline__ unsigned ordkey(float f) {
  unsigned u = __float_as_uint(f);
  return (u & 0x80000000u) ? ~u : (u | 0x80000000u);
}

// ---------------------------------------------------------------------------
// Pass 2: one block per batch. Peak detect + top-8 (jax top_k tie-breaking:
// larger value first, lower index on ties), window scatter to byte mask,
// gather per-row stats of the M=1024 selected rows -> mean, 1/(std+eps).
// ---------------------------------------------------------------------------
__global__ __launch_bounds__(256) void k_pattern(const float* __restrict__ rowsum,
                                                 const float* __restrict__ rowsumsq,
                                                 unsigned char* __restrict__ maskbuf,
                                                 float* __restrict__ stats) {
  const int b   = blockIdx.x;
  const int tid = threadIdx.x;

  __shared__ float s_xt[T_DIM];
  __shared__ float s_pts[T_DIM];
  __shared__ unsigned long long s_key[256];
  __shared__ float s_rs[256];
  __shared__ float s_rq[256];
  __shared__ int   s_ind[NPAT];

  // Load xt (= per-row channel sums) for this batch.
  for (int i = tid; i < T_DIM; i += 256) s_xt[i] = rowsum[b * T_DIM + i];
  __syncthreads();

  // 3-window max-pool with -inf padding; keep value only at local peaks.
  for (int i = tid; i < T_DIM; i += 256) {
    const float l = (i > 0)         ? s_xt[i - 1] : -INFINITY;
    const float r = (i < T_DIM - 1) ? s_xt[i + 1] : -INFINITY;
    const float c = s_xt[i];
    const float m = fmaxf(c, fmaxf(l, r));
    s_pts[i] = (c == m) ? c : 0.0f;
  }
  __syncthreads();

  // Top-8 by (value desc, index asc): packed u64 key, 8 argmax rounds.
  for (int k = 0; k < NPAT; ++k) {
    unsigned long long best = 0ull;
    for (int i = tid; i < T_DIM; i += 256) {
      const unsigned long long key =
          ((unsigned long long)ordkey(s_pts[i]) << 32) |
          (unsigned)(T_DIM - 1 - i);
      if (key > best) best = key;
    }
    s_key[tid] = best;
    __syncthreads();
    for (int off = 128; off > 0; off >>= 1) {
      if (tid < off && s_key[tid + off] > s_key[tid]) s_key[tid] = s_key[tid + off];
      __syncthreads();
    }
    if (tid == 0) {
      const int idx = T_DIM - 1 - (int)(s_key[0] & 0xffffffffu);
      s_ind[k]   = idx;
      s_pts[idx] = -INFINITY;   // exclude from later rounds
    }
    __syncthreads();
  }

  // Zero this batch's mask row (ws is poisoned by the harness).
  for (int i = tid; i < T_DIM; i += 256) maskbuf[b * T_DIM + i] = 0;
  __syncthreads();

  // Scatter the 8 clipped windows (duplicates benign: same value written).
  for (int m = tid; m < M_TOT; m += 256) {
    const int k = m >> 7, j = m & (PLEN - 1);
    int pos = s_ind[k] + j - PLEN / 2;
    pos = min(max(pos, 0), T_DIM - 1);
    maskbuf[b * T_DIM + pos] = 1;
  }

  // Gathered-comps statistics via precomputed per-row sums (duplicates kept,
  // matching take_along_axis semantics).
  float ps = 0.f, pq = 0.f;
  for (int m = tid; m < M_TOT; m += 256) {
    const int k = m >> 7, j = m & (PLEN - 1);
    int pos = s_ind[k] + j - PLEN / 2;
    pos = min(max(pos, 0), T_DIM - 1);
    ps += rowsum[b * T_DIM + pos];
    pq += rowsumsq[b * T_DIM + pos];
  }
  s_rs[tid] = ps; s_rq[tid] = pq;
  __syncthreads();
  for (int off = 128; off > 0; off >>= 1) {
    if (tid < off) { s_rs[tid] += s_rs[tid + off]; s_rq[tid] += s_rq[tid + off]; }
    __syncthreads();
  }
  if (tid == 0) {
    const float N    = (float)M_TOT * (float)D_FEAT;   // 1048576
    const float S    = s_rs[0], Q = s_rq[0];
    const float mean = S / N;
    float var = (Q - S * S / N) / (N - 1.0f);          // ddof=1 (torch.std)
    var = fmaxf(var, 0.0f);
    stats[2 * b]     = mean;
    stats[2 * b + 1] = 1.0f / (sqrtf(var) + 1e-8f);
  }
}

// ---------------------------------------------------------------------------
// Pass 3: y = (gamma*(x-mean)*scale + beta) * sigmoid_mask. Fully coalesced
// float4 in / float4 out.
// ---------------------------------------------------------------------------
__global__ __launch_bounds__(256) void k_finalize(const float* __restrict__ x,
                                                  const float* __restrict__ gamma,
                                                  const float* __restrict__ beta,
                                                  const unsigned char* __restrict__ maskbuf,
                                                  const float* __restrict__ stats,
                                                  float* __restrict__ out) {
  const int rt  = blockIdx.x;           // b*T + t
  const int b   = rt >> 12;
  const int tid = threadIdx.x;
  const float mean  = stats[2 * b];
  const float scale = stats[2 * b + 1];
  const float mv = maskbuf[rt] ? 0.8807970779778823f   // sigmoid(2)
                               : 0.04742587317756678f; // sigmoid(-3)
  const size_t base = (size_t)rt * D_FEAT;
  const float4 v = ((const float4*)(x + base))[tid];
  const float4 g = ((const float4*)gamma)[tid];
  const float4 e = ((const float4*)beta)[tid];
  float4 o;
  o.x = (g.x * (v.x - mean) * scale + e.x) * mv;
  o.y = (g.y * (v.y - mean) * scale + e.y) * mv;
  o.z = (g.z * (v.z - mean) * scale + e.z) * mv;
  o.w = (g.w * (v.w - mean) * scale + e.w) * mv;
  ((float4*)(out + base))[tid] = o;
}

// ---------------------------------------------------------------------------
extern "C" void kernel_launch(void* const* d_in, const int* in_sizes, int n_in,
                              void* d_out, int out_size, void* d_ws, size_t ws_size,
                              hipStream_t stream) {
  const float* x     = (const float*)d_in[0];
  const float* gamma = (const float*)d_in[1];
  const float* beta  = (const float*)d_in[2];
  float*       out   = (float*)d_out;

  // Workspace layout (~576 KiB):
  float*         rowsum   = (float*)d_ws;                       // B*T floats
  float*         rowsumsq = rowsum + B_DIM * T_DIM;             // B*T floats
  unsigned char* maskbuf  = (unsigned char*)(rowsumsq + B_DIM * T_DIM); // B*T bytes
  float*         stats    = (float*)(maskbuf + B_DIM * T_DIM);  // 2*B floats

  const int rows = B_DIM * T_DIM;  // 65536

  k_rowstats<<<rows, 256, 0, stream>>>(x, rowsum, rowsumsq);
  k_pattern <<<B_DIM, 256, 0, stream>>>(rowsum, rowsumsq, maskbuf, stats);
  k_finalize<<<rows, 256, 0, stream>>>(x, gamma, beta, maskbuf, stats, out);
}